// NoGradEncoding_55061480735341
// MI455X (gfx1250) — compile-verified
//
#include <hip/hip_runtime.h>
#include <stdint.h>

// Instant-NGP HashGrid encoding for MI455X (gfx1250, wave32).
// L=16 levels, F=2 feats, T=2^19 entries/level, N=1M points, out [N, 32] f32.
//
// Gather/latency-bound: 1 GB of data-dependent 8B gathers served from the
// 192MB L2 (table = 64MB, L2-resident). WMMA is inapplicable (no shared
// matrix operand). CDNA5-specific paths used:
//   - global_load_async_to_lds_b128 + s_wait_asynccnt: coalesced x staging
//   - non-temporal b128 output stores: keep the 128MB write stream from
//     evicting the gather-serving table out of L2
//   - pre-shifted XOR hashes: corner byte-offsets in 3 VALU ops each,
//     saddr(SGPR base) + 32-bit VGPR byte-offset gather addressing

#define HG_LEVELS 16
#define HG_T      (1u << 19)
#define HG_MASK8  ((HG_T - 1u) << 3)     // byte-offset mask (float2 = 8B)
#define HG_P1     2654435761u
#define HG_P2     805459861u

typedef float v4f __attribute__((ext_vector_type(4)));   // true clang vector: valid for nontemporal builtins

__global__ __launch_bounds__(256)
void hashgrid_encode_kernel(const float* __restrict__ x,
                            const float* __restrict__ table,
                            float* __restrict__ out) {
    // floor(16 * 1.4472692012786865^l) for l = 0..15 (f32 math in reference)
    constexpr float kRes[HG_LEVELS] = {
        16.f, 23.f, 33.f, 48.f, 70.f, 101.f, 147.f, 212.f,
        307.f, 445.f, 645.f, 933.f, 1351.f, 1955.f, 2830.f, 4095.f
    };

    __shared__ float sx[768];  // 256 points * 3 coords = 3072 bytes

    const int tid = threadIdx.x;
    const long long pointBase = (long long)blockIdx.x * 256;

    // ---- Async stage of the x tile into LDS (gfx1250 async copy engine) ----
    // 3072 bytes = 192 lanes * 16B, perfectly coalesced, tracked by ASYNCcnt.
    {
        uint32_t ldsBase = (uint32_t)(uintptr_t)(void*)sx;  // low 32 bits = group-relative LDS offset
        if (tid < 192) {
            uint32_t ldsAddr = ldsBase + (uint32_t)tid * 16u;
            uint64_t gaddr = (uint64_t)(uintptr_t)(x + pointBase * 3) + (uint32_t)tid * 16u;
            asm volatile("global_load_async_to_lds_b128 %0, %1, off"
                         :: "v"(ldsAddr), "v"(gaddr)
                         : "memory");
        }
        asm volatile("s_wait_asynccnt 0x0" ::: "memory");
        __syncthreads();
    }

    const float x0 = sx[tid * 3 + 0];
    const float x1 = sx[tid * 3 + 1];
    const float x2 = sx[tid * 3 + 2];

    float outv[2 * HG_LEVELS];

#pragma unroll
    for (int l = 0; l < HG_LEVELS; ++l) {
        // uniform per-level base (SGPR pair after unroll)
        const char* __restrict__ tl = (const char*)table + (size_t)l * (HG_T * 8u);
        const float res = kRes[l];

        const float px = x0 * res, py = x1 * res, pz = x2 * res;
        const float fx = floorf(px), fy = floorf(py), fz = floorf(pz);
        const float rx = px - fx, ry = py - fy, rz = pz - fz;

        const uint32_t ix = (uint32_t)fx, iy = (uint32_t)fy, iz = (uint32_t)fz;

        // Pre-shifted hash partials: ((a^b^c)&MASK)<<3 == (a<<3 ^ b<<3 ^ c<<3) & (MASK<<3)
        // => each corner's BYTE offset is xor,xor,and — no per-gather shift.
        const uint32_t hx0 = ix << 3,              hx1 = hx0 + 8u;
        const uint32_t hy0 = (iy * HG_P1) << 3,    hy1 = hy0 + (HG_P1 << 3);
        const uint32_t hz0 = (iz * HG_P2) << 3,    hz1 = hz0 + (HG_P2 << 3);

        const uint32_t o0 = (hx0 ^ hy0 ^ hz0) & HG_MASK8;
        const uint32_t o1 = (hx1 ^ hy0 ^ hz0) & HG_MASK8;
        const uint32_t o2 = (hx0 ^ hy1 ^ hz0) & HG_MASK8;
        const uint32_t o3 = (hx1 ^ hy1 ^ hz0) & HG_MASK8;
        const uint32_t o4 = (hx0 ^ hy0 ^ hz1) & HG_MASK8;
        const uint32_t o5 = (hx1 ^ hy0 ^ hz1) & HG_MASK8;
        const uint32_t o6 = (hx0 ^ hy1 ^ hz1) & HG_MASK8;
        const uint32_t o7 = (hx1 ^ hy1 ^ hz1) & HG_MASK8;

        // 8 independent 8-byte gathers in flight (global_load_b64, L2-resident)
        const float2 f0 = *(const float2*)(tl + o0);
        const float2 f1 = *(const float2*)(tl + o1);
        const float2 f2 = *(const float2*)(tl + o2);
        const float2 f3 = *(const float2*)(tl + o3);
        const float2 f4 = *(const float2*)(tl + o4);
        const float2 f5 = *(const float2*)(tl + o5);
        const float2 f6 = *(const float2*)(tl + o6);
        const float2 f7 = *(const float2*)(tl + o7);

        const float wx1 = rx, wx0 = 1.f - rx;
        const float wy1 = ry, wy0 = 1.f - ry;
        const float wz1 = rz, wz0 = 1.f - rz;
        const float wy0z0 = wy0 * wz0, wy1z0 = wy1 * wz0;
        const float wy0z1 = wy0 * wz1, wy1z1 = wy1 * wz1;
        const float w0 = wx0 * wy0z0, w1 = wx1 * wy0z0;
        const float w2 = wx0 * wy1z0, w3 = wx1 * wy1z0;
        const float w4 = wx0 * wy0z1, w5 = wx1 * wy0z1;
        const float w6 = wx0 * wy1z1, w7 = wx1 * wy1z1;

        float a = w0 * f0.x;
        float b = w0 * f0.y;
        a = fmaf(w1, f1.x, a); b = fmaf(w1, f1.y, b);
        a = fmaf(w2, f2.x, a); b = fmaf(w2, f2.y, b);
        a = fmaf(w3, f3.x, a); b = fmaf(w3, f3.y, b);
        a = fmaf(w4, f4.x, a); b = fmaf(w4, f4.y, b);
        a = fmaf(w5, f5.x, a); b = fmaf(w5, f5.y, b);
        a = fmaf(w6, f6.x, a); b = fmaf(w6, f6.y, b);
        a = fmaf(w7, f7.x, a); b = fmaf(w7, f7.y, b);

        outv[2 * l]     = a;
        outv[2 * l + 1] = b;
    }

    // Each lane writes its full contiguous 128B row as 8 x b128 NT stores:
    // the output is write-once/never-read, so the non-temporal hint keeps
    // the hash table resident in L2 for the gathers.
    v4f* __restrict__ orow = (v4f*)(out + (size_t)(pointBase + tid) * (2 * HG_LEVELS));
#pragma unroll
    for (int k = 0; k < 8; ++k) {
        v4f v = { outv[4 * k + 0], outv[4 * k + 1], outv[4 * k + 2], outv[4 * k + 3] };
        __builtin_nontemporal_store(v, &orow[k]);
    }
}

extern "C" void kernel_launch(void* const* d_in, const int* in_sizes, int n_in,
                              void* d_out, int out_size, void* d_ws, size_t ws_size,
                              hipStream_t stream) {
    (void)n_in; (void)out_size; (void)d_ws; (void)ws_size;
    const float* x     = (const float*)d_in[0];   // [N, 3] f32
    const float* table = (const float*)d_in[1];   // [16, 2^19, 2] f32
    float* out         = (float*)d_out;           // [N, 32] f32

    const int npts   = in_sizes[0] / 3;           // 1048576
    const int blocks = npts / 256;                // 4096, exact

    hipLaunchKernelGGL(hashgrid_encode_kernel, dim3(blocks), dim3(256), 0, stream,
                       x, table, out);
}